// NCEAverage_86723979641240
// MI455X (gfx1250) — compile-verified
//
#include <hip/hip_runtime.h>
#include <math.h>

typedef __attribute__((ext_vector_type(2))) float v2f;
typedef __attribute__((ext_vector_type(8))) float v8f;

#define WAVES_PER_BLOCK 8
#define BLOCK_THREADS 256

// ---------------------------------------------------------------------------
// Pass 1: gathered dot products via V_WMMA_F32_16X16X4_F32 (full f32 precision),
// exp(logit/T) -> d_out (unnormalized), per-block deterministic partial sums.
// One wave = one batch row b x one tile of 16 candidates.
// ---------------------------------------------------------------------------
__global__ __launch_bounds__(BLOCK_THREADS) void nce_pass1(
    const float* __restrict__ x, const int* __restrict__ y,
    const float* __restrict__ memory, const int* __restrict__ idx,
    const float* __restrict__ params, float* __restrict__ out,
    float* __restrict__ partials, int Kp1, int tilesPerB, int totalTiles)
{
    constexpr int DIMV = 128;
    __shared__ float red[BLOCK_THREADS];

    const int tid  = threadIdx.x;
    const int lane = tid & 31;
    const int wave = tid >> 5;
    const int tile = blockIdx.x * WAVES_PER_BLOCK + wave;

    float val = 0.0f;
    if (tile < totalTiles) {                 // wave-uniform guard (last block may hold idle waves)
        const int b     = tile / tilesPerB;
        const int t     = tile - b * tilesPerB;
        const int kbase = t * 16;
        const float T   = params[1];

        // A-matrix row index for this lane (candidate kbase + (lane&15)).
        // Candidate 0 of each batch row uses y[b] (idx[:,0] := y, without
        // mutating the input). Clamp out-of-range candidates for the load.
        int kk  = kbase + (lane & 15);
        int kkc = (kk < Kp1) ? kk : (Kp1 - 1);
        int row = (kkc == 0) ? y[b] : idx[b * Kp1 + kkc];

        const float* __restrict__ mrow = memory + (long)row * DIMV;
        const float* __restrict__ xrow = x + b * DIMV;

        // Per ISA A(16x4) layout: lanes 0-15 -> K={0,1}, lanes 16-31 -> K={2,3}
        const int koff = (lane >> 4) << 1;

        v8f acc = {};
        #pragma unroll
        for (int c = 0; c < DIMV / 4; ++c) {
            const int k = 4 * c + koff;
            v2f a, bb;
            a.x  = mrow[k];     a.y  = mrow[k + 1];   // gathered rows (A)
            bb.x = xrow[k];     bb.y = xrow[k + 1];   // x[b] broadcast over N (B)
            acc = __builtin_amdgcn_wmma_f32_16x16x4_f32(
                false, a, false, bb, (short)0, acc, false, false);
        }

        // All 16 N-columns of D are identical (B columns replicated), so
        // acc[r] on lane l holds dot(candidate 8*(l>=16)+r).  Lanes 0-7 take
        // candidates 0-7, lanes 16-23 take candidates 8-15.
        const int sel = lane & 7;
        float dotv = acc[0];
        #pragma unroll
        for (int r = 1; r < 8; ++r) dotv = (sel == r) ? acc[r] : dotv;

        const bool active = ((lane & 8) == 0);
        const int  m      = (lane & 7) + ((lane >> 4) << 3);
        const int  kkw    = kbase + m;
        if (active && kkw < Kp1) {
            float eo = expf(dotv / T);
            out[b * Kp1 + kkw] = eo;
            val = eo;
        }
    }

    // Deterministic fixed-order block reduction of eo contributions.
    red[tid] = val;
    __syncthreads();
    #pragma unroll
    for (int s = BLOCK_THREADS / 2; s > 0; s >>= 1) {
        if (tid < s) red[tid] += red[tid + s];
        __syncthreads();
    }
    if (tid == 0) partials[blockIdx.x] = red[0];
}

// ---------------------------------------------------------------------------
// Pass 2: single block reduces block partials in a fixed order and computes
// invZ = 1/Z with Z = (Z0 < 0) ? mean(eo)*outputSize : Z0.
// ---------------------------------------------------------------------------
__global__ __launch_bounds__(BLOCK_THREADS) void nce_pass2(
    const float* __restrict__ partials, const float* __restrict__ params,
    float* __restrict__ zslot, int nPartials, float invCount, float outputSize)
{
    __shared__ float red[BLOCK_THREADS];
    const int tid = threadIdx.x;
    float s = 0.0f;
    for (int i = tid; i < nPartials; i += BLOCK_THREADS) s += partials[i];
    red[tid] = s;
    __syncthreads();
    #pragma unroll
    for (int st = BLOCK_THREADS / 2; st > 0; st >>= 1) {
        if (tid < st) red[tid] += red[tid + st];
        __syncthreads();
    }
    if (tid == 0) {
        float Z0    = params[2];
        float total = red[0];
        float Z     = (Z0 < 0.0f) ? total * invCount * outputSize : Z0;
        zslot[0] = 1.0f / Z;
    }
}

// ---------------------------------------------------------------------------
// Pass 3: scale d_out by invZ (grid-stride).
// ---------------------------------------------------------------------------
__global__ void nce_pass3(float* __restrict__ out,
                          const float* __restrict__ zslot, int n)
{
    const float invZ = zslot[0];
    for (int i = blockIdx.x * blockDim.x + threadIdx.x; i < n;
         i += gridDim.x * blockDim.x)
        out[i] *= invZ;
}

extern "C" void kernel_launch(void* const* d_in, const int* in_sizes, int n_in,
                              void* d_out, int out_size, void* d_ws, size_t ws_size,
                              hipStream_t stream)
{
    const float* x      = (const float*)d_in[0];   // [B, D] f32
    const int*   y      = (const int*)  d_in[1];   // [B]    int
    const float* memory = (const float*)d_in[2];   // [N, D] f32
    const int*   idx    = (const int*)  d_in[3];   // [B, K+1] int
    const float* params = (const float*)d_in[4];   // [4]    f32
    float*       out    = (float*)d_out;           // [B, K+1] f32

    const int  B     = in_sizes[1];
    const int  D     = 128;
    const int  Kp1   = in_sizes[3] / B;
    const long Nrows = (long)(in_sizes[2] / D);

    const int tilesPerB  = (Kp1 + 15) / 16;
    const int totalTiles = B * tilesPerB;
    const int grid1      = (totalTiles + WAVES_PER_BLOCK - 1) / WAVES_PER_BLOCK;

    float* partials = (float*)d_ws;       // grid1 floats (all written each call)
    float* zslot    = partials + grid1;   // 1 float

    nce_pass1<<<grid1, BLOCK_THREADS, 0, stream>>>(
        x, y, memory, idx, params, out, partials, Kp1, tilesPerB, totalTiles);

    const float invCount = 1.0f / ((float)B * (float)Kp1);
    nce_pass2<<<1, BLOCK_THREADS, 0, stream>>>(
        partials, params, zslot, grid1, invCount, (float)Nrows);

    int grid3 = (out_size + BLOCK_THREADS - 1) / BLOCK_THREADS;
    if (grid3 > 4096) grid3 = 4096;
    nce_pass3<<<grid3, BLOCK_THREADS, 0, stream>>>(out, zslot, out_size);
}